// Ansatz_fb_56908316672071
// MI455X (gfx1250) — compile-verified
//
#include <hip/hip_runtime.h>
#include <math.h>

// ---------------------------------------------------------------------------
// FermiNet-style ansatz forward for MI455X (gfx1250).
// All dense math runs through V_WMMA_F32_16X16X4_F32 (fp32 WMMA) — the network
// is tiny (~0.5 GFLOP, ~8MB weights, fully L2-resident), so the goal is fused
// wave-per-tile WMMA GEMMs in full fp32 (slogdet is precision-sensitive).
// ---------------------------------------------------------------------------

#define N_U 64
#define N_D 64
#define NE  128
#define N_A 32
#define NSV 512
#define NPV 64

typedef __attribute__((ext_vector_type(2))) float v2f;
typedef __attribute__((ext_vector_type(8))) float v8f;

// ---------------- geometry features ----------------------------------------
__global__ void geom_sa_kernel(const float* __restrict__ r, const float* __restrict__ a,
                               float* __restrict__ s_v, float* __restrict__ env) {
    int e = blockIdx.x * blockDim.x + threadIdx.x;
    if (e >= NE) return;
    float rx = r[e * 3 + 0], ry = r[e * 3 + 1], rz = r[e * 3 + 2];
    float esum = 0.f;
    for (int at = 0; at < N_A; ++at) {
        float dx = rx - a[at * 3 + 0];
        float dy = ry - a[at * 3 + 1];
        float dz = rz - a[at * 3 + 2];
        float len = sqrtf(dx * dx + dy * dy + dz * dz);
        s_v[e * (4 * N_A) + 4 * at + 0] = dx;
        s_v[e * (4 * N_A) + 4 * at + 1] = dy;
        s_v[e * (4 * N_A) + 4 * at + 2] = dz;
        s_v[e * (4 * N_A) + 4 * at + 3] = len;
        esum += expf(-len);
    }
    env[e] = esum;
}

__global__ void geom_pp_kernel(const float* __restrict__ r, float* __restrict__ p_v) {
    int t = blockIdx.x * blockDim.x + threadIdx.x;
    if (t >= NE * NE) return;
    int i = t / NE, j = t % NE;
    float dx = r[j * 3 + 0] - r[i * 3 + 0];
    float dy = r[j * 3 + 1] - r[i * 3 + 1];
    float dz = r[j * 3 + 2] - r[i * 3 + 2];
    float len = (i == j) ? 0.f : sqrtf(dx * dx + dy * dy + dz * dz);
    p_v[t * 4 + 0] = dx;
    p_v[t * 4 + 1] = dy;
    p_v[t * 4 + 2] = dz;
    p_v[t * 4 + 3] = len;
}

// ---------------- fb_block helpers -----------------------------------------
// means layout: [su(f_s)][sd(f_s)][pu(NE*f_p)][pd(NE*f_p)]
__global__ void means_kernel(const float* __restrict__ s_v, int f_s,
                             const float* __restrict__ p_v, int f_p,
                             float* __restrict__ means) {
    int t = blockIdx.x * blockDim.x + threadIdx.x;
    int n_s = f_s, n_p = NE * f_p;
    if (t >= 2 * n_s + 2 * n_p) return;
    if (t < n_s) {
        float s = 0.f;
        for (int e = 0; e < N_U; ++e) s += s_v[e * f_s + t];
        means[t] = s * (1.0f / N_U);
    } else if (t < 2 * n_s) {
        int c = t - n_s;
        float s = 0.f;
        for (int e = N_U; e < NE; ++e) s += s_v[e * f_s + c];
        means[t] = s * (1.0f / N_D);
    } else if (t < 2 * n_s + n_p) {
        int idx = t - 2 * n_s;
        int j = idx / f_p, c = idx % f_p;
        float s = 0.f;
        for (int i = 0; i < N_U; ++i) s += p_v[(i * NE + j) * f_p + c];
        means[t] = s * (1.0f / N_U);
    } else {
        int idx = t - 2 * n_s - n_p;
        int j = idx / f_p, c = idx % f_p;
        float s = 0.f;
        for (int i = N_U; i < NE; ++i) s += p_v[(i * NE + j) * f_p + c];
        means[t] = s * (1.0f / N_D);
    }
}

__global__ void concat_kernel(const float* __restrict__ s_v, int f_s,
                              const float* __restrict__ means, int f_p,
                              float* __restrict__ s_in) {
    int fin = 3 * f_s + 2 * f_p;
    int t = blockIdx.x * blockDim.x + threadIdx.x;
    if (t >= NE * fin) return;
    int e = t / fin, c = t % fin;
    const float* su = means;
    const float* sd = means + f_s;
    const float* pu = means + 2 * f_s;
    const float* pd = pu + NE * f_p;
    float v;
    if (c < f_s)                 v = su[c];
    else if (c < 2 * f_s)        v = sd[c - f_s];
    else if (c < 2 * f_s + f_p)  v = pu[e * f_p + (c - 2 * f_s)];
    else if (c < 2 * f_s + 2 * f_p) v = pd[e * f_p + (c - 2 * f_s - f_p)];
    else                         v = s_v[e * f_s + (c - 2 * f_s - 2 * f_p)];
    s_in[t] = v;
}

// ---------------- fp32 WMMA GEMM: C = [tanh](A@B + bias) [+ resid] ----------
// One wave per 16x16 C tile; K-loop of V_WMMA_F32_16X16X4_F32.
// Requires M,N % 16 == 0 and K % 4 == 0 (true for every GEMM here).
__global__ void wmma_gemm_kernel(const float* __restrict__ A, int lda,
                                 const float* __restrict__ B, int ldb,
                                 const float* __restrict__ bias,
                                 const float* __restrict__ resid,
                                 float* __restrict__ C, int ldc,
                                 int M, int N, int K, int do_tanh) {
    int wave   = blockIdx.x * (blockDim.x >> 5) + (threadIdx.x >> 5);
    int tilesN = N >> 4;
    int tiles  = (M >> 4) * tilesN;
    if (wave >= tiles) return;
    int tm = (wave / tilesN) << 4;
    int tn = (wave % tilesN) << 4;
    int lane = threadIdx.x & 31;
    int half = lane >> 4;     // 0: lanes 0-15, 1: lanes 16-31
    int mn   = lane & 15;

    v8f acc = {0.f, 0.f, 0.f, 0.f, 0.f, 0.f, 0.f, 0.f};
    const float* Ap = A + (size_t)(tm + mn) * lda;   // A row for this lane
    const float* Bp = B + tn + mn;                   // B column for this lane
    for (int k0 = 0; k0 < K; k0 += 4) {
        int ka = k0 + 2 * half;  // lanes 0-15: K=k0,k0+1 ; lanes 16-31: K=k0+2,k0+3
        v2f av, bv;
        av.x = Ap[ka];
        av.y = Ap[ka + 1];
        bv.x = Bp[(size_t)ka * ldb];
        bv.y = Bp[(size_t)(ka + 1) * ldb];
        acc = __builtin_amdgcn_wmma_f32_16x16x4_f32(
            /*neg_a=*/false, av, /*neg_b=*/false, bv,
            /*c_mod=*/(short)0, acc, /*reuse_a=*/false, /*reuse_b=*/false);
    }

    int col = tn + mn;
    float b = bias ? bias[col] : 0.f;
#pragma unroll
    for (int i = 0; i < 8; ++i) {
        int row = tm + half * 8 + i;   // C/D layout: VGPR i -> M=i (lanes 0-15), M=8+i (16-31)
        float v = acc[i] + b;
        if (do_tanh) v = tanhf(v);
        if (resid) v += resid[(size_t)row * ldc + col];
        C[(size_t)row * ldc + col] = v;
    }
}

// ---------------- envelope scaling ------------------------------------------
__global__ void scale_kernel(float* __restrict__ wu, float* __restrict__ wd,
                             const float* __restrict__ env) {
    int t = blockIdx.x * blockDim.x + threadIdx.x;
    if (t >= 64 * 64) return;
    int i = t >> 6;                 // row index
    wu[t] *= env[i];                // up electrons 0..63
    wd[t] *= env[64 + i];           // down electrons 64..127
}

// ---------------- slogdet of the two 64x64 matrices -------------------------
// LU with partial pivoting in LDS, one block of 64 threads handles both.
// With a single det-pair, log|psi| + maxlogdet == ld_u + ld_d exactly.
__global__ void slogdet_kernel(const float* __restrict__ wu,
                               const float* __restrict__ wd,
                               float* __restrict__ out) {
    __shared__ float Msh[64 * 64];
    __shared__ int   s_piv;
    __shared__ float s_acc;
    int t = threadIdx.x;  // 64 threads
    if (t == 0) s_acc = 0.f;
    for (int m = 0; m < 2; ++m) {
        const float* src = (m == 0) ? wu : wd;
        for (int j = 0; j < 64; ++j) Msh[t * 64 + j] = src[t * 64 + j];
        __syncthreads();
        for (int k = 0; k < 64; ++k) {
            if (t == 0) {
                int p = k;
                float best = fabsf(Msh[k * 64 + k]);
                for (int i = k + 1; i < 64; ++i) {
                    float v = fabsf(Msh[i * 64 + k]);
                    if (v > best) { best = v; p = i; }
                }
                s_piv = p;
            }
            __syncthreads();
            int p = s_piv;
            if (p != k) {  // swap rows k and p; each thread owns one column
                float tmp = Msh[k * 64 + t];
                Msh[k * 64 + t] = Msh[p * 64 + t];
                Msh[p * 64 + t] = tmp;
            }
            __syncthreads();
            if (t > k) {
                float f = Msh[t * 64 + k] / Msh[k * 64 + k];
                for (int j = k; j < 64; ++j) Msh[t * 64 + j] -= f * Msh[k * 64 + j];
            }
            __syncthreads();
        }
        if (t == 0) {
            float ld = 0.f;
            for (int k = 0; k < 64; ++k) ld += logf(fabsf(Msh[k * 64 + k]));
            s_acc += ld;
        }
        __syncthreads();
    }
    if (t == 0) out[0] = s_acc;
}

// ---------------- host orchestration ----------------------------------------
extern "C" void kernel_launch(void* const* d_in, const int* in_sizes, int n_in,
                              void* d_out, int out_size, void* d_ws, size_t ws_size,
                              hipStream_t stream) {
    (void)in_sizes; (void)n_in; (void)out_size; (void)ws_size;
    const float* r       = (const float*)d_in[0];
    const float* atoms   = (const float*)d_in[1];
    const float* Vw[3]   = {(const float*)d_in[2], (const float*)d_in[6],  (const float*)d_in[10]};
    const float* Vb[3]   = {(const float*)d_in[3], (const float*)d_in[7],  (const float*)d_in[11]};
    const float* Ww[3]   = {(const float*)d_in[4], (const float*)d_in[8],  (const float*)d_in[12]};
    const float* Wb[3]   = {(const float*)d_in[5], (const float*)d_in[9],  (const float*)d_in[13]};
    const float* after_w = (const float*)d_in[14];
    const float* after_b = (const float*)d_in[15];
    const float* vhu_w   = (const float*)d_in[16];
    const float* vhu_b   = (const float*)d_in[17];
    const float* vhd_w   = (const float*)d_in[18];
    const float* vhd_b   = (const float*)d_in[19];
    const float* wu_w    = (const float*)d_in[20];
    const float* wu_b    = (const float*)d_in[21];
    const float* wd_w    = (const float*)d_in[22];
    const float* wd_b    = (const float*)d_in[23];

    float* w = (float*)d_ws;
    // workspace layout (float offsets)
    const size_t SV_A  = 0;                         // 128*512
    const size_t SV_B  = SV_A + (size_t)NE * NSV;   // 128*512
    const size_t P_A   = SV_B + (size_t)NE * NSV;   // 128*128*64
    const size_t P_B   = P_A + (size_t)NE * NE * NPV;
    const size_t SIN   = P_B + (size_t)NE * NE * NPV;      // 128*1664
    const size_t MEANS = SIN + (size_t)NE * 1664;          // 2*512 + 2*128*64
    const size_t ENV   = MEANS + 2 * NSV + 2 * NE * NPV;   // 128
    const size_t HU    = ENV + NE;                         // 64*256
    const size_t HD    = HU + 64 * 256;                    // 64*256
    const size_t WU    = HD + 64 * 256;                    // 64*64
    const size_t WD    = WU + 64 * 64;                     // 64*64

    auto gemm = [&](const float* A, int lda, const float* B, int ldb,
                    const float* bias, const float* resid,
                    float* C, int ldc, int M, int N, int K, int do_tanh) {
        int tiles  = (M / 16) * (N / 16);
        int wpb    = 4;                       // 4 waves (128 threads) per block
        int blocks = (tiles + wpb - 1) / wpb;
        wmma_gemm_kernel<<<blocks, wpb * 32, 0, stream>>>(
            A, lda, B, ldb, bias, resid, C, ldc, M, N, K, do_tanh);
    };

    // 1) geometry features + envelope sums
    geom_sa_kernel<<<1, 128, 0, stream>>>(r, atoms, w + SV_A, w + ENV);
    geom_pp_kernel<<<(NE * NE + 255) / 256, 256, 0, stream>>>(r, w + P_A);

    // 2) three feature-block layers
    float* s_cur = w + SV_A; float* s_nxt = w + SV_B;
    float* p_cur = w + P_A;  float* p_nxt = w + P_B;
    int f_s = 4 * N_A, f_p = 4;
    for (int l = 0; l < 3; ++l) {
        int fin = 3 * f_s + 2 * f_p;
        int mt = 2 * f_s + 2 * NE * f_p;
        means_kernel<<<(mt + 255) / 256, 256, 0, stream>>>(s_cur, f_s, p_cur, f_p, w + MEANS);
        int ct = NE * fin;
        concat_kernel<<<(ct + 255) / 256, 256, 0, stream>>>(s_cur, f_s, w + MEANS, f_p, w + SIN);
        // s_v = tanh(s_in @ Vw + Vb)
        gemm(w + SIN, fin, Vw[l], NSV, Vb[l], nullptr, s_nxt, NSV, NE, NSV, fin, 1);
        // p_v = tanh(p_v @ Ww + Wb) [+ p_v if input width == N_PV]
        const float* resid = (f_p == NPV) ? p_cur : nullptr;
        gemm(p_cur, f_p, Ww[l], NPV, Wb[l], resid, p_nxt, NPV, NE * NE, NPV, f_p, 1);
        { float* t = s_cur; s_cur = s_nxt; s_nxt = t; }
        { float* t = p_cur; p_cur = p_nxt; p_nxt = t; }
        f_s = NSV; f_p = NPV;
    }

    // 3) "after" block: s_v = tanh(fb_block(s_v, p_v) @ after_w + after_b)
    {
        int fin = 3 * f_s + 2 * f_p;  // 1664
        int mt = 2 * f_s + 2 * NE * f_p;
        means_kernel<<<(mt + 255) / 256, 256, 0, stream>>>(s_cur, f_s, p_cur, f_p, w + MEANS);
        int ct = NE * fin;
        concat_kernel<<<(ct + 255) / 256, 256, 0, stream>>>(s_cur, f_s, w + MEANS, f_p, w + SIN);
        gemm(w + SIN, fin, after_w, NSV, after_b, nullptr, s_nxt, NSV, NE, NSV, fin, 1);
        { float* t = s_cur; s_cur = s_nxt; s_nxt = t; }
    }

    // 4) heads: hu/hd = s_v[u/d] @ vh*_w + b ; orb = h @ w*_w + b
    gemm(s_cur,                 NSV, vhu_w, 256, vhu_b, nullptr, w + HU, 256, 64, 256, NSV, 0);
    gemm(s_cur + (size_t)N_U * NSV, NSV, vhd_w, 256, vhd_b, nullptr, w + HD, 256, 64, 256, NSV, 0);
    gemm(w + HU, 256, wu_w, 64, wu_b, nullptr, w + WU, 64, 64, 64, 256, 0);
    gemm(w + HD, 256, wd_w, 64, wd_b, nullptr, w + WD, 64, 64, 64, 256, 0);

    // 5) envelope scaling + slogdet combine
    scale_kernel<<<16, 256, 0, stream>>>(w + WU, w + WD, w + ENV);
    slogdet_kernel<<<1, 64, 0, stream>>>(w + WU, w + WD, (float*)d_out);
}